// MSSSIM_75290776699231
// MI455X (gfx1250) — compile-verified
//
#include <hip/hip_runtime.h>
#include <hip/hip_bf16.h>
#include <math.h>

typedef __attribute__((ext_vector_type(2))) float v2f;
typedef __attribute__((ext_vector_type(8))) float v8f;

#define NC_TOT 48            // N*C = 16*3
#define WS 11                // gaussian window size

__device__ __forceinline__ float map_val(float xv, float yv, int m) {
    switch (m) {
        case 0: return xv;
        case 1: return yv;
        case 2: return xv * xv;
        case 3: return yv * yv;
        default: return xv * yv;
    }
}

// One block = 256 threads = 8 waves. Block computes a 16-row x 128-col output
// tile of the VALID 11x11 depthwise conv for one (n,c) image, for all 5 maps,
// then accumulates SSIM/CS sums. Each wave owns a 16x16 output sub-tile.
// Separable conv as banded-matrix WMMA (V_WMMA_F32_16X16X4_F32, K 26->28).
// launch_bounds(256,1): give the allocator the full VGPR file so the
// compiler's multi-chain WMMA pipelining doesn't spill to scratch.
__global__ void __launch_bounds__(256, 1)
ssim_scale_kernel(const float* __restrict__ X,
                  const float* __restrict__ Y,
                  int H, int W,
                  const float* __restrict__ win,
                  float* __restrict__ ssim_sum,
                  float* __restrict__ cs_sum,
                  int scale_idx) {
    __shared__ float xs[32][144];     // input tile rows 0..31, cols 0..139 used
    __shared__ float ys[32][144];
    __shared__ float hs[8][32][16];   // per-wave hconv scratch (rows 0..31)
    __shared__ float band[64];        // zero-padded 1-D gaussian: band[16+d]=g[d]
    __shared__ float blk_ss[8], blk_cs[8];

    const int tid  = threadIdx.x;
    const int lane = tid & 31;
    const int wave = tid >> 5;
    const int half = lane >> 4;       // 0: lanes 0-15, 1: lanes 16-31
    const int l16  = lane & 15;

    const int Hout = H - 10;
    const int Wout = W - 10;
    const int R0 = blockIdx.y * 16;
    const int C0 = blockIdx.x * 128;
    const long base = (long)blockIdx.z * H * W;

    // Zero-padded band table: band[16+d] = g[d] for d in [0,10], else 0.
    // g recovered from 2-D window (k2d = g (x) g): g[d] = k2d[d][5]/sqrt(k2d[5][5]).
    if (tid < 64) {
        int d = tid - 16;
        float v = 0.0f;
        if (d >= 0 && d < 11) v = win[d * WS + 5] / sqrtf(win[5 * WS + 5]);
        band[tid] = v;
    }

    // Stage input tile (zero-fill out of bounds so everything stays finite).
    for (int i = tid; i < 32 * 140; i += 256) {
        int r = i / 140, c = i % 140;
        int gr = R0 + r, gc = C0 + c;
        float xv = 0.0f, yv = 0.0f;
        if (gr < H && gc < W) {
            long o = base + (long)gr * W + gc;
            xv = X[o];
            yv = Y[o];
        }
        xs[r][c] = xv;
        ys[r][c] = yv;
    }
    __syncthreads();

    // Hoisted banded fragments: identical for horizontal-B and vertical-A,
    // invariant across maps and row groups. 7 chunks x v2f, kept in VGPRs.
    v2f gfrag[7];
    #pragma unroll
    for (int q = 0; q < 7; ++q) {
        int idx = 16 + 4 * q + 2 * half - l16;   // in [1, 43]
        gfrag[q].x = band[idx];
        gfrag[q].y = band[idx + 1];
    }

    const int cb = wave * 16;         // column base of this wave's sub-tile

    v8f res[5];
    #pragma unroll
    for (int m = 0; m < 5; ++m) {
        // ---- horizontal pass: two 16-row groups cover hconv rows 0..31 ----
        #pragma unroll
        for (int grp = 0; grp < 2; ++grp) {
            v8f acc = {};
            const int row = grp * 16 + l16;   // A-matrix row (data row)
            const float* xr = &xs[row][cb];
            const float* yr = &ys[row][cb];
            #pragma unroll
            for (int q = 0; q < 7; ++q) {
                const int k0 = 4 * q + 2 * half;           // even -> 8B aligned
                v2f xv = *(const v2f*)(xr + k0);
                v2f yv = *(const v2f*)(yr + k0);
                v2f a;
                a.x = map_val(xv.x, yv.x, m);
                a.y = map_val(xv.y, yv.y, m);
                acc = __builtin_amdgcn_wmma_f32_16x16x4_f32(
                    false, a, false, gfrag[q], (short)0, acc, false, false);
            }
            #pragma unroll
            for (int e = 0; e < 8; ++e)
                hs[wave][grp * 16 + e + 8 * half][l16] = acc[e];
        }

        // ---- vertical pass: Out = G(16x28) x hconv(28x16) ----
        {
            v8f acc = {};
            #pragma unroll
            for (int q = 0; q < 7; ++q) {
                const int k0 = 4 * q + 2 * half;
                v2f b;
                b.x = hs[wave][k0][l16];
                b.y = hs[wave][k0 + 1][l16];
                acc = __builtin_amdgcn_wmma_f32_16x16x4_f32(
                    false, gfrag[q], false, b, (short)0, acc, false, false);
            }
            res[m] = acc;
        }
    }

    // ---- per-element SSIM / CS, masked at tile edges ----
    const float c1 = 1.0e-4f;   // (0.01*1)^2
    const float c2 = 9.0e-4f;   // (0.03*1)^2
    float ssim_acc = 0.0f, cs_acc = 0.0f;
    #pragma unroll
    for (int e = 0; e < 8; ++e) {
        int gr = R0 + e + 8 * half;      // C/D layout: VGPR e -> M = e + 8*half
        int gc = C0 + cb + l16;          // N = lane % 16
        float mx  = res[0][e], my  = res[1][e];
        float exx = res[2][e], eyy = res[3][e], exy = res[4][e];
        float mxx = mx * mx, myy = my * my, mxy = mx * my;
        float sx = exx - mxx, sy = eyy - myy, sxyv = exy - mxy;
        float cs = (2.0f * sxyv + c2) / (sx + sy + c2);
        float ss = (2.0f * mxy + c1) / (mxx + myy + c1) * cs;
        float msk = (gr < Hout && gc < Wout) ? 1.0f : 0.0f;
        ssim_acc += msk * ss;
        cs_acc   += msk * cs;
    }

    // wave32 reduction, then block reduction, then one atomic pair per block
    #pragma unroll
    for (int off = 16; off > 0; off >>= 1) {
        ssim_acc += __shfl_xor(ssim_acc, off);
        cs_acc   += __shfl_xor(cs_acc, off);
    }
    if (lane == 0) {
        blk_ss[wave] = ssim_acc;
        blk_cs[wave] = cs_acc;
    }
    __syncthreads();
    if (tid == 0) {
        float s = 0.0f, c = 0.0f;
        #pragma unroll
        for (int w = 0; w < 8; ++w) { s += blk_ss[w]; c += blk_cs[w]; }
        atomicAdd(&ssim_sum[scale_idx * NC_TOT + blockIdx.z], s);
        atomicAdd(&cs_sum[scale_idx * NC_TOT + blockIdx.z],   c);
    }
}

// 2x2 average pool, stride 2 (H, W even at every scale here).
// 3D grid: x = output-col tile, y = output row, z = image -> no int div/mod.
__global__ void avgpool2_kernel(const float* __restrict__ in,
                                float* __restrict__ out,
                                int H, int W) {
    int Ho = H >> 1, Wo = W >> 1;
    int wo = blockIdx.x * blockDim.x + threadIdx.x;
    int ho = blockIdx.y;
    int img = blockIdx.z;
    if (wo >= Wo) return;
    const float* p = in + (long)img * H * W + (long)(2 * ho) * W + 2 * wo;
    out[(long)img * Ho * Wo + (long)ho * Wo + wo] =
        0.25f * (p[0] + p[1] + p[W] + p[W + 1]);
}

__global__ void zero_kernel(float* p, int n) {
    int i = blockIdx.x * blockDim.x + threadIdx.x;
    if (i < n) p[i] = 0.0f;
}

// Combine: prod over scales of relu(cs_mean)^w (scales 0..3) * ssim_mean^w4, mean over N*C.
__global__ void finalize_kernel(const float* __restrict__ ssim_sum,
                                const float* __restrict__ cs_sum,
                                const float* __restrict__ wts,
                                float* __restrict__ out, int H0) {
    __shared__ float red[64];
    int t = threadIdx.x;
    float v = 0.0f;
    if (t < NC_TOT) {
        v = 1.0f;
        #pragma unroll
        for (int s = 0; s < 4; ++s) {
            int hs = H0 >> s;
            float cnt = (float)((hs - 10) * (hs - 10));
            float cm = cs_sum[s * NC_TOT + t] / cnt;
            v *= powf(fmaxf(cm, 0.0f), wts[s]);
        }
        int hs4 = H0 >> 4;
        float cnt4 = (float)((hs4 - 10) * (hs4 - 10));
        float sm = ssim_sum[4 * NC_TOT + t] / cnt4;
        v *= powf(sm, wts[4]);   // no relu on last-scale ssim (matches reference)
    }
    red[t] = v;
    __syncthreads();
    for (int off = 32; off > 0; off >>= 1) {
        if (t < off) red[t] += red[t + off];
        __syncthreads();
    }
    if (t == 0) out[0] = red[0] / (float)NC_TOT;
}

extern "C" void kernel_launch(void* const* d_in, const int* in_sizes, int n_in,
                              void* d_out, int out_size, void* d_ws, size_t ws_size,
                              hipStream_t stream) {
    const float* X   = (const float*)d_in[0];   // (16,3,512,512)
    const float* Yp  = (const float*)d_in[1];
    const float* win = (const float*)d_in[2];   // (3,1,11,11) -> use channel 0
    const float* wts = (const float*)d_in[3];   // (5,)
    float* out = (float*)d_out;

    float* ws = (float*)d_ws;
    float* ssim_sum = ws;                 // 5*48 floats
    float* cs_sum   = ws + 5 * NC_TOT;    // 5*48 floats
    float* pyr      = ws + 10 * NC_TOT;

    const long n1 = (long)NC_TOT * 256 * 256;
    const long n2 = (long)NC_TOT * 128 * 128;
    const long n3 = (long)NC_TOT * 64 * 64;
    const long n4 = (long)NC_TOT * 32 * 32;
    float* x1 = pyr;      float* y1 = x1 + n1;
    float* x2 = y1 + n1;  float* y2 = x2 + n2;
    float* x3 = y2 + n2;  float* y3 = x3 + n3;
    float* x4 = y3 + n3;  float* y4 = x4 + n4;

    zero_kernel<<<2, 256, 0, stream>>>(ws, 10 * NC_TOT);

    auto launch_scale = [&](const float* xp, const float* yp, int Hs, int s) {
        int HoS = Hs - 10;
        dim3 grid((HoS + 127) / 128, (HoS + 15) / 16, NC_TOT);
        ssim_scale_kernel<<<grid, 256, 0, stream>>>(xp, yp, Hs, Hs, win,
                                                    ssim_sum, cs_sum, s);
    };
    auto pool = [&](const float* in, float* o, int Hs) {
        int Wo = Hs / 2;
        int bx = (Wo + 255) / 256;
        dim3 grid(bx, Wo, NC_TOT);    // Ho == Wo (square images)
        avgpool2_kernel<<<grid, 256, 0, stream>>>(in, o, Hs, Hs);
    };

    launch_scale(X, Yp, 512, 0);
    pool(X, x1, 512);  pool(Yp, y1, 512);
    launch_scale(x1, y1, 256, 1);
    pool(x1, x2, 256); pool(y1, y2, 256);
    launch_scale(x2, y2, 128, 2);
    pool(x2, x3, 128); pool(y2, y3, 128);
    launch_scale(x3, y3, 64, 3);
    pool(x3, x4, 64);  pool(y3, y4, 64);
    launch_scale(x4, y4, 32, 4);

    finalize_kernel<<<1, 64, 0, stream>>>(ssim_sum, cs_sum, wts, out, 512);
}